// GATBlock_343597384054
// MI455X (gfx1250) — compile-verified
//
#include <hip/hip_runtime.h>
#include <hip/hip_bf16.h>
#include <math.h>

// ---------------- problem constants ----------------
#define N_NODES   50000
#define N_EDGES   800000
#define E_TOT     (N_EDGES + N_NODES)   // self-loops appended
#define DIM       128
#define HEADS     4
#define HEAD_DIM  32
#define NEG_SLOPE 0.2f
#define LN_EPS    1e-5f

typedef __attribute__((ext_vector_type(2))) float v2f;
typedef __attribute__((ext_vector_type(8))) float v8f;

// ---------------- helpers ----------------
// monotone float <-> uint encoding so float max == unsigned max
__device__ __forceinline__ unsigned enc_f32(float f) {
  unsigned u = __float_as_uint(f);
  return (u & 0x80000000u) ? ~u : (u | 0x80000000u);
}
__device__ __forceinline__ float dec_f32(unsigned k) {
  return (k & 0x80000000u) ? __uint_as_float(k ^ 0x80000000u)
                           : __uint_as_float(~k);
}
__device__ __forceinline__ float lrelu(float x) {
  return x > 0.f ? x : NEG_SLOPE * x;
}

// ---------------- K0: init accumulators ----------------
__global__ __launch_bounds__(256) void k_init(float* __restrict__ out,
                                              unsigned* __restrict__ emax,
                                              float* __restrict__ denom) {
  int tid = blockIdx.x * blockDim.x + threadIdx.x;
  if (tid < N_NODES * DIM)   out[tid] = 0.f;
  if (tid < N_NODES * HEADS) { emax[tid] = 0u; denom[tid] = 0.f; }
}

// ---------------- K1: h = x @ W via f32 WMMA ----------------
// One wave computes a 16(M) x 128(N) strip; K looped in steps of 4.
// A 16x4 f32 fragment: lanes 0-15 -> M=lane, {K=k0,k0+1}; lanes 16-31 -> {K=k0+2,k0+3}
// B 4x16 fragment mirrors C/D striping: v0 = rows {k0, k0+2}, v1 = rows {k0+1, k0+3}
__global__ __launch_bounds__(256) void k_gemm(const float* __restrict__ X,
                                              const float* __restrict__ W,
                                              float* __restrict__ H) {
  const int wv   = blockIdx.x * 8 + (threadIdx.x >> 5);
  const int lane = threadIdx.x & 31;
  const int m0   = wv * 16;
  if (m0 >= N_NODES) return;              // uniform per wave: EXEC stays all-1s
  const int half = lane >> 4;             // 0 or 1
  const int l16  = lane & 15;

  v8f acc[8] = {};
  const float* __restrict__ arow = X + (size_t)(m0 + l16) * DIM;

  for (int k0 = 0; k0 < DIM; k0 += 4) {
    const int ka = k0 + half * 2;
    v2f a;
    a.x = arow[ka];
    a.y = arow[ka + 1];
#pragma unroll
    for (int nt = 0; nt < 8; ++nt) {
      v2f b;
      b.x = W[(size_t)ka * DIM + nt * 16 + l16];
      b.y = W[(size_t)(ka + 1) * DIM + nt * 16 + l16];
      acc[nt] = __builtin_amdgcn_wmma_f32_16x16x4_f32(
          /*neg_a=*/false, a, /*neg_b=*/false, b,
          /*c_mod=*/(short)0, acc[nt], /*reuse_a=*/false, /*reuse_b=*/false);
    }
  }
  // C/D layout: VGPR r -> rows {m0+r (lanes 0-15), m0+8+r (lanes 16-31)}, col = l16
#pragma unroll
  for (int nt = 0; nt < 8; ++nt) {
#pragma unroll
    for (int r = 0; r < 8; ++r) {
      H[(size_t)(m0 + half * 8 + r) * DIM + nt * 16 + l16] = acc[nt][r];
    }
  }
}

// ---------------- K2: per-(node,head) attention logits ----------------
__global__ __launch_bounds__(256) void k_attn(const float* __restrict__ H,
                                              const float* __restrict__ att_src,
                                              const float* __restrict__ att_dst,
                                              float* __restrict__ a_s,
                                              float* __restrict__ a_d) {
  int tid = blockIdx.x * blockDim.x + threadIdx.x;
  if (tid >= N_NODES * HEADS) return;
  const int node = tid >> 2;
  const int hd   = tid & 3;
  const float* __restrict__ hp = H + (size_t)node * DIM + hd * HEAD_DIM;
  const float* __restrict__ sp = att_src + hd * HEAD_DIM;
  const float* __restrict__ dp = att_dst + hd * HEAD_DIM;
  float ss = 0.f, sd = 0.f;
#pragma unroll 8
  for (int d = 0; d < HEAD_DIM; ++d) {
    float hv = hp[d];
    ss = fmaf(hv, sp[d], ss);
    sd = fmaf(hv, dp[d], sd);
  }
  a_s[tid] = ss;
  a_d[tid] = sd;
}

// ---------------- K3: segment max (hardware u32 atomic max) ----------------
__global__ __launch_bounds__(256) void k_edge_max(const int* __restrict__ ei,
                                                  const float* __restrict__ a_s,
                                                  const float* __restrict__ a_d,
                                                  unsigned* __restrict__ emax) {
  int tid = blockIdx.x * blockDim.x + threadIdx.x;
  if (tid >= E_TOT * HEADS) return;
  const int e  = tid >> 2;
  const int hd = tid & 3;
  int src, dst;
  if (e < N_EDGES) { src = ei[e]; dst = ei[N_EDGES + e]; }
  else             { src = dst = e - N_EDGES; }           // self-loop
  const float v = lrelu(a_s[src * HEADS + hd] + a_d[dst * HEADS + hd]);
  atomicMax(&emax[dst * HEADS + hd], enc_f32(v));
}

// ---------------- K4: exp + segment sum (hardware f32 atomic add) --------
__global__ __launch_bounds__(256) void k_edge_sum(const int* __restrict__ ei,
                                                  const float* __restrict__ a_s,
                                                  const float* __restrict__ a_d,
                                                  const unsigned* __restrict__ emax,
                                                  float* __restrict__ denom) {
  int tid = blockIdx.x * blockDim.x + threadIdx.x;
  if (tid >= E_TOT * HEADS) return;
  const int e  = tid >> 2;
  const int hd = tid & 3;
  int src, dst;
  if (e < N_EDGES) { src = ei[e]; dst = ei[N_EDGES + e]; }
  else             { src = dst = e - N_EDGES; }
  const int seg = dst * HEADS + hd;
  const float v  = lrelu(a_s[src * HEADS + hd] + a_d[dst * HEADS + hd]);
  const float ex = expf(v - dec_f32(emax[seg]));
  unsafeAtomicAdd(&denom[seg], ex);
}

// ---------------- K5: weighted message scatter-add ----------------
// one wave32 per edge; lane l handles features [4l, 4l+4) -> head = l>>3
__global__ __launch_bounds__(256) void k_edge_msg(const int* __restrict__ ei,
                                                  const float* __restrict__ H,
                                                  const float* __restrict__ a_s,
                                                  const float* __restrict__ a_d,
                                                  const unsigned* __restrict__ emax,
                                                  const float* __restrict__ denom,
                                                  float* __restrict__ out) {
  const int wv = blockIdx.x * 8 + (threadIdx.x >> 5);
  if (wv >= E_TOT) return;
  const int lane = threadIdx.x & 31;
  int src, dst;
  if (wv < N_EDGES) { src = ei[wv]; dst = ei[N_EDGES + wv]; }
  else              { src = dst = wv - N_EDGES; }
  const int hd  = lane >> 3;
  const int seg = dst * HEADS + hd;
  const float v     = lrelu(a_s[src * HEADS + hd] + a_d[dst * HEADS + hd]);
  const float alpha = expf(v - dec_f32(emax[seg])) / (denom[seg] + 1e-16f);

  const int f = lane * 4;
  const float4 hv = *(const float4*)(H + (size_t)src * DIM + f);
  float* __restrict__ op = out + (size_t)dst * DIM + f;
  unsafeAtomicAdd(op + 0, hv.x * alpha);
  unsafeAtomicAdd(op + 1, hv.y * alpha);
  unsafeAtomicAdd(op + 2, hv.z * alpha);
  unsafeAtomicAdd(op + 3, hv.w * alpha);
}

// ---------------- K6: bias + LayerNorm + exact GELU (in-place) ----------
__global__ __launch_bounds__(256) void k_epilogue(float* __restrict__ out,
                                                  const float* __restrict__ bias,
                                                  const float* __restrict__ gamma,
                                                  const float* __restrict__ beta) {
  const int wv = blockIdx.x * 8 + (threadIdx.x >> 5);
  if (wv >= N_NODES) return;
  const int lane = threadIdx.x & 31;
  const int f = lane * 4;
  float* __restrict__ row = out + (size_t)wv * DIM + f;

  float4 v  = *(float4*)row;
  const float4 bb = *(const float4*)(bias + f);
  v.x += bb.x; v.y += bb.y; v.z += bb.z; v.w += bb.w;

  float s = v.x + v.y + v.z + v.w;
#pragma unroll
  for (int o = 16; o > 0; o >>= 1) s += __shfl_xor(s, o, 32);
  const float mu = s * (1.f / DIM);

  const float d0 = v.x - mu, d1 = v.y - mu, d2 = v.z - mu, d3 = v.w - mu;
  float q = d0 * d0 + d1 * d1 + d2 * d2 + d3 * d3;
#pragma unroll
  for (int o = 16; o > 0; o >>= 1) q += __shfl_xor(q, o, 32);
  const float rstd = rsqrtf(q * (1.f / DIM) + LN_EPS);

  const float4 g  = *(const float4*)(gamma + f);
  const float4 bt = *(const float4*)(beta + f);
  float y0 = d0 * rstd * g.x + bt.x;
  float y1 = d1 * rstd * g.y + bt.y;
  float y2 = d2 * rstd * g.z + bt.z;
  float y3 = d3 * rstd * g.w + bt.w;

  const float k = 0.70710678118654752f;  // 1/sqrt(2)
  float4 r;
  r.x = 0.5f * y0 * (1.f + erff(y0 * k));
  r.y = 0.5f * y1 * (1.f + erff(y1 * k));
  r.z = 0.5f * y2 * (1.f + erff(y2 * k));
  r.w = 0.5f * y3 * (1.f + erff(y3 * k));
  *(float4*)row = r;
}

// ---------------- host launcher ----------------
extern "C" void kernel_launch(void* const* d_in, const int* in_sizes, int n_in,
                              void* d_out, int out_size, void* d_ws, size_t ws_size,
                              hipStream_t stream) {
  const float* x       = (const float*)d_in[0];
  const int*   ei      = (const int*)d_in[1];     // [2, E] flat, int32
  const float* W       = (const float*)d_in[2];
  const float* att_src = (const float*)d_in[3];
  const float* att_dst = (const float*)d_in[4];
  const float* bias    = (const float*)d_in[5];
  const float* gamma   = (const float*)d_in[6];
  const float* beta    = (const float*)d_in[7];
  float* out = (float*)d_out;

  // workspace carve-up (16B aligned): h | a_s | a_d | emax | denom  (~28.8 MB)
  char* ws = (char*)d_ws;
  float*    H     = (float*)(ws);
  float*    a_s   = (float*)(ws + 25600000);      // 50000*128*4
  float*    a_d   = (float*)(ws + 26400000);
  unsigned* emax  = (unsigned*)(ws + 27200000);
  float*    denom = (float*)(ws + 28000000);

  k_init<<<(N_NODES * DIM) / 256, 256, 0, stream>>>(out, emax, denom);
  k_gemm<<<(N_NODES / 16 + 7) / 8, 256, 0, stream>>>(x, W, H);
  k_attn<<<(N_NODES * HEADS + 255) / 256, 256, 0, stream>>>(H, att_src, att_dst, a_s, a_d);
  k_edge_max<<<(E_TOT * HEADS + 255) / 256, 256, 0, stream>>>(ei, a_s, a_d, emax);
  k_edge_sum<<<(E_TOT * HEADS + 255) / 256, 256, 0, stream>>>(ei, a_s, a_d, emax, denom);
  k_edge_msg<<<(E_TOT + 7) / 8, 256, 0, stream>>>(ei, H, a_s, a_d, emax, denom, out);
  k_epilogue<<<(N_NODES + 7) / 8, 256, 0, stream>>>(out, bias, gamma, beta);
}